// BlockMLP_24111946399890
// MI455X (gfx1250) — compile-verified
//
#include <hip/hip_runtime.h>

// ---------------------------------------------------------------------------
// BlockMLP forward for MI455X (gfx1250, wave32).
//
// Roofline: x is [8,50,93000,2] fp32 = 297.6 MB read once -> ~12.8 us at
// 23.3 TB/s HBM. Compute is ~150 MFLOP (negligible). Bandwidth-shaped
// streaming reduction; fp32 throughout.
//
// Structure exploit: every CHR_LENS[c] is divisible by 100, so the "ragged"
// segment-sum is uniform per chromosome with block length bl[c] = L_c/100.
// seg_ids is ignored.
//
// Kernel 1: grid = B*N*n_chr = 4000 WGs x 256 threads (8 waves).
//   - Each wave reduces blocks q = wave, wave+8, ...: lanes stream float2
//     (both ploidies of one marker, global_load_b64) * w1, prefetch the next
//     block (global_prefetch), shfl_xor reduce, relu(+b1)*w2 -> padded LDS.
//   - Layer 2 (dot of 100 per (p,c)) on wave 0 as 25 chained
//     V_WMMA_F32_16X16X4_F32: A rows = p from the zero-padded LDS tile
//     (branchless ds_loads), B = all-ones so B's VGPR striping is irrelevant;
//     D[p][0] accumulates the dot. relu(+b2) -> outs scratch in d_ws.
// Kernel 2: 8 WGs (one per batch b): actor max over p, critic mean.
// ---------------------------------------------------------------------------

typedef float v2f __attribute__((ext_vector_type(2)));
typedef float v8f __attribute__((ext_vector_type(8)));

#define NCHR 10
#define NBPC 100           // blocks per chromosome
#define MTOT 93000         // sum of CHR_LENS
#define NBN  400           // B*N
#define NPLO 2
#define YPAD 16            // padded row width of the layer-2 LDS tile

__device__ __constant__ int kCS[NCHR] = {0, 12000, 23000, 33500, 43500,
                                         53000, 62000, 70500, 78500, 86000};
__device__ __constant__ int kBL[NCHR] = {120, 110, 105, 100, 95, 90, 85, 80, 75, 70};

__global__ __launch_bounds__(256) void k_block_mlp(
    const float* __restrict__ x,   // [B,N,M,P] fp32
    const float* __restrict__ w1,  // [M]
    const float* __restrict__ b1,  // [1000]
    const float* __restrict__ w2,  // [10,100]
    const float* __restrict__ b2,  // [10]
    float* __restrict__ outs)      // [B*N, P, NCHR] scratch
{
  const int c  = blockIdx.x % NCHR;   // chromosome
  const int bn = blockIdx.x / NCHR;   // fused (b,n)
  const int bl = kBL[c];
  const int cs = kCS[c];

  // Zero-padded tile: ysw[q][p] = relu(block_sum + b1) * w2[c][q] for p<2,
  // 0 for p in 2..15.  Lets the WMMA A-build be branchless ds_loads.
  __shared__ float ysw[NBPC * YPAD];

  const int wave = threadIdx.x >> 5;
  const int lane = threadIdx.x & 31;

  for (int i = threadIdx.x; i < NBPC * YPAD; i += 256) ysw[i] = 0.f;
  __syncthreads();

  // Contiguous chromosome slice for this (b,n): float2 = both p of a marker.
  const float2* __restrict__ xc  = (const float2*)x + (size_t)bn * MTOT + cs;
  const float*  __restrict__ w1c = w1 + cs;
  const float*  __restrict__ w2c = w2 + c * NBPC;

  // ---- Layer 1: weighted per-block reduction (the bandwidth-bound part) ----
  for (int q = wave; q < NBPC; q += 8) {
    // Prefetch this wave's next block (<= 960 B): 32 lanes x 32 B stride
    // covers 1024 B.  Lowers to global_prefetch_b8.
    if (q + 8 < NBPC) {
      const float2* nb = xc + (q + 8) * bl;
      __builtin_prefetch((const void*)(nb + (lane << 2)), 0, 0);
    }

    const float2* __restrict__ xb = xc + q * bl;
    const float*  __restrict__ wb = w1c + q * bl;
    float a0 = 0.f, a1 = 0.f;
    for (int j = lane; j < bl; j += 32) {       // bl <= 120 -> <=4 iters
      float2 v = xb[j];                         // global_load_b64, coalesced
      float  w = wb[j];                         // L2-resident
      a0 = fmaf(v.x, w, a0);
      a1 = fmaf(v.y, w, a1);
    }
#pragma unroll
    for (int off = 16; off > 0; off >>= 1) {    // wave32 cross-lane reduce
      a0 += __shfl_xor(a0, off, 32);
      a1 += __shfl_xor(a1, off, 32);
    }
    if (lane == 0) {
      const float bb = b1[c * NBPC + q];
      const float s2 = w2c[q];                  // pre-scale by layer-2 weight
      ysw[q * YPAD + 0] = fmaxf(a0 + bb, 0.f) * s2;
      ysw[q * YPAD + 1] = fmaxf(a1 + bb, 0.f) * s2;
    }
  }
  __syncthreads();

  // ---- Layer 2 on the matrix pipe: D[p][n] += sum_K A[p][K] * 1.0 ----------
  // A is the documented 16x4 f32 layout: lane l holds row M = l&15,
  // VGPR0 = K = (l<16 ? 0 : 2), VGPR1 = K+1.  Rows >= 2 read the zeroed
  // padding, so the build is branchless.  B = all ones, so B's striping
  // cannot matter; D's row sums are the w2-dots.
  if (wave == 0) {
    const int M  = lane & 15;
    const int kh = (lane >> 4) << 1;            // 0 or 2
    v8f acc = {};
    v2f ones; ones.x = 1.0f; ones.y = 1.0f;
#pragma unroll
    for (int t = 0; t < 25; ++t) {              // 25 * K=4 covers q = 0..99
      const int q0 = t * 4 + kh;
      v2f a;
      a.x = ysw[q0 * YPAD + M];
      a.y = ysw[(q0 + 1) * YPAD + M];
      acc = __builtin_amdgcn_wmma_f32_16x16x4_f32(
          /*neg_a=*/false, a, /*neg_b=*/false, ones,
          /*c_mod=*/(short)0, acc, /*reuse_a=*/false, /*reuse_b=*/false);
    }
    if (lane == 0) {
      // D layout: VGPR v, lanes 0-15 -> M=v, N=lane.  Lane 0: acc[0]=D[0][0],
      // acc[1]=D[1][0] -> the two ploidy dot products.
      const float bb = b2[c];
      float* o = outs + (size_t)bn * (NPLO * NCHR);
      o[0 * NCHR + c] = fmaxf(acc[0] + bb, 0.f);
      o[1 * NCHR + c] = fmaxf(acc[1] + bb, 0.f);
    }
  }
}

// ---- Heads: actor = max_p(outs . wa) + ba ; critic = bc + mean_{n,p} -------
__global__ __launch_bounds__(64) void k_heads(
    const float* __restrict__ outs,  // [400, 2, 10]
    const float* __restrict__ wa, const float* __restrict__ ba,
    const float* __restrict__ wc, const float* __restrict__ bc,
    float* __restrict__ out)         // [400] actor ++ [8] critic
{
  const int b = blockIdx.x;          // 0..7
  const int t = threadIdx.x;         // 0..63
  __shared__ float red[64];

  float s = 0.f;
  if (t < 50) {
    const float* o = outs + (size_t)(b * 50 + t) * (NPLO * NCHR);
    float a0 = 0.f, a1 = 0.f, cr = 0.f;
#pragma unroll
    for (int c = 0; c < NCHR; ++c) {
      a0 = fmaf(o[c],        wa[c], a0);
      a1 = fmaf(o[NCHR + c], wa[c], a1);
      cr = fmaf(o[c] + o[NCHR + c], wc[c], cr);
    }
    out[b * 50 + t] = fmaxf(a0, a1) + ba[0];
    s = cr;
  }
  red[t] = s;
  __syncthreads();
  if (t == 0) {
    float tot = 0.f;
#pragma unroll
    for (int i = 0; i < 64; ++i) tot += red[i];
    out[NBN + b] = bc[0] + tot * (1.0f / (50.0f * 2.0f));
  }
}

extern "C" void kernel_launch(void* const* d_in, const int* in_sizes, int n_in,
                              void* d_out, int out_size, void* d_ws, size_t ws_size,
                              hipStream_t stream) {
  const float* x  = (const float*)d_in[0];
  const float* w1 = (const float*)d_in[1];
  const float* b1 = (const float*)d_in[2];
  const float* w2 = (const float*)d_in[3];
  const float* b2 = (const float*)d_in[4];
  const float* wa = (const float*)d_in[5];
  const float* ba = (const float*)d_in[6];
  const float* wc = (const float*)d_in[7];
  const float* bc = (const float*)d_in[8];
  // d_in[9] = seg_ids: unused (block structure is uniform per chromosome).

  float* outs = (float*)d_ws;        // 400*2*10 floats = 32 KB scratch
  float* out  = (float*)d_out;       // 408 floats

  k_block_mlp<<<dim3(NBN * NCHR), dim3(256), 0, stream>>>(x, w1, b1, w2, b2, outs);
  k_heads<<<dim3(8), dim3(64), 0, stream>>>(outs, wa, ba, wc, bc, out);
}